// KalmanFilter_82240033783883
// MI455X (gfx1250) — compile-verified
//
#include <hip/hip_runtime.h>
#include <math.h>

// Problem constants (match reference)
#define G_ 1024
#define T_ 512
#define S_ 64
#define M_ 8
#define LDA 65   // padded LDS row stride (floats) -> conflict-free row AND column access

typedef __attribute__((ext_vector_type(2))) float v2f;
typedef __attribute__((ext_vector_type(8))) float v8f;

// D(64x64) = A(64x64) * op(B)(64x64) [+ epilogue], computed with
// V_WMMA_F32_16X16X4_F32.  8 waves -> 16 tiles of 16x16, 2 tiles/wave.
// MODE 0: D = A*B            (TB=false)
// MODE 1: D = A*B^T + KR*K^T (Joseph-form extra term, 8-wide dot in epilogue)
// MODE 2: D = A*B^T + Q
template<bool TB, int MODE>
__device__ __forceinline__ void mm64(const float* __restrict__ A,
                                     const float* __restrict__ B,
                                     float* __restrict__ D,
                                     const float* __restrict__ sKR,
                                     const float* __restrict__ sK,
                                     const float* __restrict__ sQ,
                                     int tid)
{
    const int lane = tid & 31;
    const int wave = tid >> 5;
    const int lm   = lane & 15;
    const int khi  = (lane >> 4) << 1;   // 0 (lanes 0-15) or 2 (lanes 16-31)
#pragma unroll
    for (int rep = 0; rep < 2; ++rep) {
        const int tile = wave + (rep << 3);     // 0..15
        const int ti = tile >> 2;
        const int tj = tile & 3;
        const int mrow = ti * 16 + lm;          // A-row this lane supplies
        const int ncol = tj * 16 + lm;          // B/D-column this lane supplies
        v8f acc = {};
#pragma unroll
        for (int k0 = 0; k0 < S_; k0 += 4) {
            const int k = k0 + khi;
            v2f a, b;
            // A 16x4 layout: VGPR v holds K = 2*(lane/16) + v for row M = lane%16
            a.x = A[mrow * LDA + k];
            a.y = A[mrow * LDA + k + 1];
            // B 4x16 layout: VGPR v holds row K = 2*(lane/16) + v, col N = lane%16
            if (TB) {
                b.x = B[ncol * LDA + k];        // B^T(k, n) = B(n, k)
                b.y = B[ncol * LDA + k + 1];
            } else {
                b.x = B[(k)     * LDA + ncol];
                b.y = B[(k + 1) * LDA + ncol];
            }
            acc = __builtin_amdgcn_wmma_f32_16x16x4_f32(
                      false, a, false, b, (short)0, acc, false, false);
        }
        // C/D layout: VGPR v -> row = tile_row + v + 8*(lane/16), col = lane%16
        const int rbase = ti * 16 + ((lane >> 4) << 3);
#pragma unroll
        for (int v = 0; v < 8; ++v) {
            const int r = rbase + v;
            float out = acc[v];
            if (MODE == 1) {
                float add = 0.f;
#pragma unroll
                for (int i = 0; i < M_; ++i)
                    add += sKR[r * M_ + i] * sK[ncol * M_ + i];
                out += add;
            } else if (MODE == 2) {
                out += sQ[r * LDA + ncol];
            }
            D[r * LDA + ncol] = out;
        }
    }
}

__global__ __launch_bounds__(256)
void kalman_wmma_kernel(const float* __restrict__ y,    // [G,T,M]
                        const float* __restrict__ F,    // [S,S]
                        const float* __restrict__ H,    // [M,S]
                        const float* __restrict__ Q,    // [S,S]
                        const float* __restrict__ R,    // [M,M]
                        const float* __restrict__ m0,   // [G,S]
                        const float* __restrict__ P0,   // [S,S]
                        float* __restrict__ out_mean,   // [G,T,M]
                        float* __restrict__ out_cov)    // [G,T,M,M]
{
    __shared__ float sP[S_ * LDA];   // state covariance (prediction)
    __shared__ float sA[S_ * LDA];   // ImKH / scratch
    __shared__ float sB[S_ * LDA];   // matmul temporary
    __shared__ float sF[S_ * LDA];
    __shared__ float sQ[S_ * LDA];
    __shared__ float sH[M_ * S_];
    __shared__ float sHP[M_ * S_];   // H*P
    __shared__ float sK[S_ * M_];    // Kalman gain
    __shared__ float sKR[S_ * M_];   // K*R
    __shared__ float sR[M_ * M_];
    __shared__ float sS[M_ * M_];    // innovation covariance
    __shared__ float sL[M_ * M_];    // its Cholesky factor
    __shared__ float sm[S_];         // state mean (prediction)
    __shared__ float sresid[M_];

    const int tid = threadIdx.x;
    const int g   = blockIdx.x;

    for (int i = tid; i < S_ * S_; i += 256) {
        const int r = i >> 6, c = i & 63;
        sF[r * LDA + c] = F[i];
        sQ[r * LDA + c] = Q[i];
        sP[r * LDA + c] = P0[i];
    }
    for (int i = tid; i < M_ * S_; i += 256) sH[i] = H[i];
    if (tid < M_ * M_) sR[tid] = R[tid];
    if (tid < S_)      sm[tid] = m0[(size_t)g * S_ + tid];
    __syncthreads();

    const float* yg  = y        + (size_t)g * T_ * M_;
    float*       omg = out_mean + (size_t)g * T_ * M_;
    float*       ocg = out_cov  + (size_t)g * T_ * M_ * M_;

    for (int t = 0; t < T_; ++t) {
        // ---- phase 1: HP = H*P  (8x64), pred_mean = H*m, resid ----
        for (int idx = tid; idx < M_ * S_; idx += 256) {
            const int i = idx >> 6, j = idx & 63;
            float acc = 0.f;
            for (int s = 0; s < S_; ++s) acc += sH[i * S_ + s] * sP[s * LDA + j];
            sHP[idx] = acc;
        }
        if (tid < M_) {
            float acc = 0.f;
            for (int s = 0; s < S_; ++s) acc += sH[tid * S_ + s] * sm[s];
            omg[t * M_ + tid]  = acc;                     // output pred mean
            sresid[tid] = yg[t * M_ + tid] - acc;
        }
        __syncthreads();

        // ---- phase 2: innovation cov Sm = HP*H^T + R (8x8), output it ----
        if (tid < M_ * M_) {
            const int i = tid >> 3, j = tid & 7;
            float acc = sR[tid];
            for (int s = 0; s < S_; ++s) acc += sHP[i * S_ + s] * sH[j * S_ + s];
            sS[tid] = acc;
            ocg[(size_t)t * M_ * M_ + tid] = acc;         // output pred cov
        }
        __syncthreads();

        // ---- phase 3: 8x8 Cholesky (single thread; trivial cost) ----
        if (tid == 0) {
            for (int i = 0; i < M_; ++i)
                for (int j = 0; j <= i; ++j) {
                    float sum = sS[i * M_ + j];
                    for (int k = 0; k < j; ++k) sum -= sL[i * M_ + k] * sL[j * M_ + k];
                    if (i == j) sL[i * M_ + i] = sqrtf(fmaxf(sum, 1e-30f));
                    else        sL[i * M_ + j] = sum / sL[j * M_ + j];
                }
        }
        __syncthreads();

        // ---- phase 4: K = P H^T S^-1 (solve per state row), m_u, KR ----
        if (tid < S_) {
            float x[M_];
#pragma unroll
            for (int i = 0; i < M_; ++i) {                // forward subst
                float acc = sHP[i * S_ + tid];
                for (int k = 0; k < i; ++k) acc -= sL[i * M_ + k] * x[k];
                x[i] = acc / sL[i * M_ + i];
            }
#pragma unroll
            for (int i = M_ - 1; i >= 0; --i) {           // backward subst
                float acc = x[i];
                for (int k = i + 1; k < M_; ++k) acc -= sL[k * M_ + i] * x[k];
                x[i] = acc / sL[i * M_ + i];
            }
            float mu = sm[tid];
#pragma unroll
            for (int i = 0; i < M_; ++i) { sK[tid * M_ + i] = x[i]; mu += x[i] * sresid[i]; }
            sm[tid] = mu;                                 // updated mean m_u
#pragma unroll
            for (int i = 0; i < M_; ++i) {                // KR = K*R
                float acc = 0.f;
#pragma unroll
                for (int j = 0; j < M_; ++j) acc += x[j] * sR[j * M_ + i];
                sKR[tid * M_ + i] = acc;
            }
        }
        __syncthreads();

        // ---- phase 5: ImKH = I - K*H  -> sA ----
        for (int idx = tid; idx < S_ * S_; idx += 256) {
            const int s = idx >> 6, n = idx & 63;
            float acc = (s == n) ? 1.f : 0.f;
#pragma unroll
            for (int i = 0; i < M_; ++i) acc -= sK[s * M_ + i] * sH[i * S_ + n];
            sA[s * LDA + n] = acc;
        }
        __syncthreads();

        // ---- phase 6: T1 = ImKH * P  (WMMA) ----
        mm64<false, 0>(sA, sP, sB, sKR, sK, sQ, tid);
        __syncthreads();

        // ---- phase 7: P_u = T1 * ImKH^T + K R K^T (WMMA + epilogue) ----
        mm64<true, 1>(sB, sA, sP, sKR, sK, sQ, tid);
        float mp = 0.f;                                   // m_p = F * m_u (overlap)
        if (tid < S_)
            for (int j = 0; j < S_; ++j) mp += sF[tid * LDA + j] * sm[j];
        __syncthreads();
        if (tid < S_) sm[tid] = mp;

        // ---- phase 8: T2 = F * P_u  (WMMA) ----
        mm64<false, 0>(sF, sP, sB, sKR, sK, sQ, tid);
        __syncthreads();

        // ---- phase 9: P_next = T2 * F^T + Q  (WMMA + epilogue) ----
        mm64<true, 2>(sB, sF, sP, sKR, sK, sQ, tid);
        __syncthreads();
    }
}

extern "C" void kernel_launch(void* const* d_in, const int* in_sizes, int n_in,
                              void* d_out, int out_size, void* d_ws, size_t ws_size,
                              hipStream_t stream) {
    const float* y  = (const float*)d_in[0];   // input [G,T,M]
    const float* F  = (const float*)d_in[1];
    const float* H  = (const float*)d_in[2];
    const float* Q  = (const float*)d_in[3];
    const float* R  = (const float*)d_in[4];
    const float* m0 = (const float*)d_in[5];
    const float* P0 = (const float*)d_in[6];
    float* out_mean = (float*)d_out;                               // [G,T,M]
    float* out_cov  = out_mean + (size_t)G_ * T_ * M_;             // [G,T,M,M]

    hipLaunchKernelGGL(kalman_wmma_kernel, dim3(G_), dim3(256), 0, stream,
                       y, F, H, Q, R, m0, P0, out_mean, out_cov);
}